// ObstacleHead_46823733461252
// MI455X (gfx1250) — compile-verified
//
#include <hip/hip_runtime.h>
#include <hip/hip_bf16.h>
#include <math.h>

// Problem constants (from reference)
#define BB 64
#define NN 16
#define HH 224
#define WWI 224
#define HW (HH * WWI)
#define HW4 (HW / 4)
#define DD 1024
#define DMM 512

typedef float v2f __attribute__((ext_vector_type(2)));
typedef float v4f __attribute__((ext_vector_type(4)));
typedef float v8f __attribute__((ext_vector_type(8)));

// ---------------------------------------------------------------------------
// Kernel: per-batch sum of target mask T (B blocks), b128 loads
// ---------------------------------------------------------------------------
__global__ void k_tsum(const float* __restrict__ T, float* __restrict__ Tsum) {
  __shared__ float red[256];
  int b = blockIdx.x;
  const v4f* t4 = (const v4f*)(T + (size_t)b * HW);
  float s = 0.f;
  for (int i = threadIdx.x; i < HW4; i += 256) { v4f v = t4[i]; s += (v.x + v.y) + (v.z + v.w); }
  red[threadIdx.x] = s; __syncthreads();
  for (int off = 128; off > 0; off >>= 1) {
    if (threadIdx.x < off) red[threadIdx.x] += red[threadIdx.x + off];
    __syncthreads();
  }
  if (threadIdx.x == 0) Tsum[b] = red[0];
}

// ---------------------------------------------------------------------------
// Kernel: overlap[b,n] = sum(om*T), mask_sum[b,n] = sum(om)   (B*N blocks)
// overlap written directly into edge[b, n*2 + 0].  b128 loads.
// ---------------------------------------------------------------------------
__global__ void k_overlap(const float* __restrict__ om, const float* __restrict__ T,
                          float* __restrict__ edge, float* __restrict__ mask_sum) {
  __shared__ float r1[256], r2[256];
  int bn = blockIdx.x;          // b*16 + n
  int b  = bn >> 4;
  const v4f* o4 = (const v4f*)(om + (size_t)bn * HW);
  const v4f* t4 = (const v4f*)(T  + (size_t)b  * HW);
  float so = 0.f, sm = 0.f;
  for (int i = threadIdx.x; i < HW4; i += 256) {
    v4f ov = o4[i]; v4f tv = t4[i];
    so += (ov.x * tv.x + ov.y * tv.y) + (ov.z * tv.z + ov.w * tv.w);
    sm += (ov.x + ov.y) + (ov.z + ov.w);
  }
  r1[threadIdx.x] = so; r2[threadIdx.x] = sm; __syncthreads();
  for (int off = 128; off > 0; off >>= 1) {
    if (threadIdx.x < off) { r1[threadIdx.x] += r1[threadIdx.x+off]; r2[threadIdx.x] += r2[threadIdx.x+off]; }
    __syncthreads();
  }
  if (threadIdx.x == 0) {
    int n = bn & 15;
    edge[b*32 + n*2 + 0] = r1[0];
    mask_sum[bn] = r2[0];
  }
}

// ---------------------------------------------------------------------------
// Kernel: IoU of floored bbox vs target mask  (B*N blocks) -> edge[b,n*2+1]
// ---------------------------------------------------------------------------
__global__ void k_iou(const float* __restrict__ T, const float* __restrict__ bbox,
                      const float* __restrict__ Tsum, float* __restrict__ edge) {
  __shared__ float red[256];
  int bn = blockIdx.x; int b = bn >> 4; int n = bn & 15;
  float x1 = floorf(bbox[bn*4 + 0]);
  float y1 = floorf(bbox[bn*4 + 1]);
  float x2 = floorf(bbox[bn*4 + 2]);
  float y2 = floorf(bbox[bn*4 + 3]);
  int c0 = max(0, (int)x1), c1 = min(WWI, (int)x2);
  int r0 = max(0, (int)y1), r1 = min(HH,  (int)y2);
  int wbox = c1 - c0, hbox = r1 - r0;
  float s = 0.f;
  if (wbox > 0 && hbox > 0) {
    int tot = wbox * hbox;
    const float* t = T + (size_t)b * HW;
    for (int i = threadIdx.x; i < tot; i += 256) {
      int rr = r0 + i / wbox, cc = c0 + i % wbox;
      s += t[rr * WWI + cc];
    }
  }
  red[threadIdx.x] = s; __syncthreads();
  for (int off = 128; off > 0; off >>= 1) {
    if (threadIdx.x < off) red[threadIdx.x] += red[threadIdx.x+off];
    __syncthreads();
  }
  if (threadIdx.x == 0) {
    float inter  = red[0];
    float rowcnt = fmaxf(fminf(y2, (float)HH)  - fmaxf(y1, 0.f), 0.f);
    float colcnt = fmaxf(fminf(x2, (float)WWI) - fmaxf(x1, 0.f), 0.f);
    float uni = rowcnt * colcnt + Tsum[b] - inter;
    edge[b*32 + n*2 + 1] = inter / (uni + 1e-8f);
  }
}

// ---------------------------------------------------------------------------
// WMMA inner product over a K range: one wave -> 64x16 output strip (4 row
// tiles of 16, shared B fragment per k-step).  fp32 via V_WMMA_F32_16X16X4_F32.
// A frag: lanes 0-15 M=lane K={k0,k0+1}; lanes 16-31 K={k0+2,k0+3}. B mirrors.
// C/D: vgpr r -> M = r + 8*(lane>=16), N = lane%16.
// ---------------------------------------------------------------------------
__device__ __forceinline__ void wmma_strip(const float* __restrict__ X,
                                           const float* __restrict__ W,
                                           int K, int Nout, int n, int l, int half,
                                           int kbeg, int kend,
                                           v8f& acc0, v8f& acc1, v8f& acc2, v8f& acc3) {
  for (int k0 = kbeg; k0 < kend; k0 += 4) {
    int ka = k0 + half * 2;
    v2f bf;
    bf.x = W[(size_t)(ka + 0) * Nout + n];
    bf.y = W[(size_t)(ka + 1) * Nout + n];
    const float* xp = X + (size_t)l * K + ka;
    v2f a0, a1, a2, a3;
    a0.x = xp[0];              a0.y = xp[1];
    a1.x = xp[(size_t)16*K];   a1.y = xp[(size_t)16*K + 1];
    a2.x = xp[(size_t)32*K];   a2.y = xp[(size_t)32*K + 1];
    a3.x = xp[(size_t)48*K];   a3.y = xp[(size_t)48*K + 1];
    acc0 = __builtin_amdgcn_wmma_f32_16x16x4_f32(false, a0, false, bf, (short)0, acc0, false, false);
    acc1 = __builtin_amdgcn_wmma_f32_16x16x4_f32(false, a1, false, bf, (short)0, acc1, false, false);
    acc2 = __builtin_amdgcn_wmma_f32_16x16x4_f32(false, a2, false, bf, (short)0, acc2, false, false);
    acc3 = __builtin_amdgcn_wmma_f32_16x16x4_f32(false, a3, false, bf, (short)0, acc3, false, false);
  }
}

// Single-pass GEMM + bias (used when K is small enough that Nout/16 waves
// have acceptable per-wave K-loop length).
__global__ void k_gemm_bias(const float* __restrict__ X, const float* __restrict__ W,
                            const float* __restrict__ bias, float* __restrict__ Y,
                            int K, int Nout) {
  int wid  = (int)(blockIdx.x * blockDim.x + threadIdx.x) >> 5;
  int lane = threadIdx.x & 31;
  int col0 = wid * 16;
  if (col0 >= Nout) return;          // wave-uniform: EXEC stays all-1s
  int half = lane >> 4;
  int l    = lane & 15;
  int n    = col0 + l;
  v8f acc0 = {}; v8f acc1 = {}; v8f acc2 = {}; v8f acc3 = {};
  wmma_strip(X, W, K, Nout, n, l, half, 0, K, acc0, acc1, acc2, acc3);
  float bv = bias[n];
#pragma unroll
  for (int r = 0; r < 8; r++) {
    int m = half * 8 + r;
    Y[(size_t)(m)      * Nout + n] = acc0[r] + bv;
    Y[(size_t)(16 + m) * Nout + n] = acc1[r] + bv;
    Y[(size_t)(32 + m) * Nout + n] = acc2[r] + bv;
    Y[(size_t)(48 + m) * Nout + n] = acc3[r] + bv;
  }
}

// Split-K partial GEMM: wave = (chunk, strip); writes partials (no bias).
__global__ void k_gemm_partial(const float* __restrict__ X, const float* __restrict__ W,
                               float* __restrict__ Yp, int K, int Nout,
                               int kchunk, int nchunks) {
  int wid  = (int)(blockIdx.x * blockDim.x + threadIdx.x) >> 5;
  int lane = threadIdx.x & 31;
  int strips = Nout >> 4;
  int strip = wid % strips;
  int chunk = wid / strips;
  if (chunk >= nchunks) return;      // wave-uniform
  int half = lane >> 4;
  int l    = lane & 15;
  int n    = strip * 16 + l;
  int kbeg = chunk * kchunk;
  int kend = min(K, kbeg + kchunk);
  v8f acc0 = {}; v8f acc1 = {}; v8f acc2 = {}; v8f acc3 = {};
  wmma_strip(X, W, K, Nout, n, l, half, kbeg, kend, acc0, acc1, acc2, acc3);
  float* yp = Yp + (size_t)chunk * 64 * Nout;
#pragma unroll
  for (int r = 0; r < 8; r++) {
    int m = half * 8 + r;
    yp[(size_t)(m)      * Nout + n] = acc0[r];
    yp[(size_t)(16 + m) * Nout + n] = acc1[r];
    yp[(size_t)(32 + m) * Nout + n] = acc2[r];
    yp[(size_t)(48 + m) * Nout + n] = acc3[r];
  }
}

// Deterministic fixed-order reduction of split-K partials + bias.
__global__ void k_reduce_bias(const float* __restrict__ Yp, const float* __restrict__ bias,
                              float* __restrict__ Y, int Nout, int nchunks) {
  int idx = (int)(blockIdx.x * 256 + threadIdx.x);
  int total = 64 * Nout;
  if (idx >= total) return;
  int n = idx % Nout;
  float s = 0.f;
  for (int c = 0; c < nchunks; c++) s += Yp[(size_t)c * total + idx];
  Y[idx] = s + bias[n];
}

// ---------------------------------------------------------------------------
// LayerNorm (biased var, eps 1e-5) + affine + ReLU, in place. 64 blocks (rows).
// ---------------------------------------------------------------------------
__global__ void k_ln_relu(float* __restrict__ Y, const float* __restrict__ g,
                          const float* __restrict__ be, int Ncols) {
  __shared__ float red[256];
  __shared__ float s_mean, s_inv;
  int row = blockIdx.x;
  float* y = Y + (size_t)row * Ncols;
  float s = 0.f;
  for (int i = threadIdx.x; i < Ncols; i += 256) s += y[i];
  red[threadIdx.x] = s; __syncthreads();
  for (int off = 128; off > 0; off >>= 1) {
    if (threadIdx.x < off) red[threadIdx.x] += red[threadIdx.x+off];
    __syncthreads();
  }
  if (threadIdx.x == 0) s_mean = red[0] / (float)Ncols;
  __syncthreads();
  float mean = s_mean;
  float v = 0.f;
  for (int i = threadIdx.x; i < Ncols; i += 256) { float d = y[i] - mean; v += d * d; }
  __syncthreads();
  red[threadIdx.x] = v; __syncthreads();
  for (int off = 128; off > 0; off >>= 1) {
    if (threadIdx.x < off) red[threadIdx.x] += red[threadIdx.x+off];
    __syncthreads();
  }
  if (threadIdx.x == 0) s_inv = 1.0f / sqrtf(red[0] / (float)Ncols + 1e-5f);
  __syncthreads();
  float inv = s_inv;
  for (int i = threadIdx.x; i < Ncols; i += 256) {
    float t = (y[i] - mean) * inv * g[i] + be[i];
    y[i] = fmaxf(t, 0.f);
  }
}

// ---------------------------------------------------------------------------
// scores[b] = (q[b] @ k[b]^T) / sqrt(512) as one 16x16x512 fp32 WMMA per wave.
// A = q rows (M=n), B = k rows as columns (N=m); both are the same strided
// row-major 16x512 access.  One wave per batch: grid 8 blocks x 8 waves.
// ---------------------------------------------------------------------------
__global__ void k_scores_wmma(const float* __restrict__ q, const float* __restrict__ kk,
                              float* __restrict__ scores) {
  int wave = (int)(blockIdx.x * blockDim.x + threadIdx.x) >> 5;   // = b, 0..63
  int lane = threadIdx.x & 31;
  int half = lane >> 4, l = lane & 15;
  const float* qb = q  + (size_t)wave * (NN*DMM);
  const float* kb = kk + (size_t)wave * (NN*DMM);
  v8f acc = {};
  for (int k0 = 0; k0 < DMM; k0 += 4) {
    int ka = k0 + half * 2;
    v2f a, b;
    a.x = qb[l*DMM + ka]; a.y = qb[l*DMM + ka + 1];
    b.x = kb[l*DMM + ka]; b.y = kb[l*DMM + ka + 1];
    acc = __builtin_amdgcn_wmma_f32_16x16x4_f32(false, a, false, b, (short)0, acc, false, false);
  }
#pragma unroll
  for (int r = 0; r < 8; r++)
    scores[wave*256 + (half*8 + r)*16 + l] = acc[r] * 0.04419417382415922f; // 1/sqrt(512)
}

// ---------------------------------------------------------------------------
// softmax over m, attn_out = P @ rel, build concat buffer
// outcat[b, n*1024 + d]       = attn_out[b,n,d]   (d < 512)
// outcat[b, n*1024 + 512 + d] = rel[b,n,d]
// ---------------------------------------------------------------------------
__global__ void k_attn(const float* __restrict__ scores, const float* __restrict__ rel,
                       float* __restrict__ outcat) {
  __shared__ float p[16][16];
  int b = blockIdx.x;
  if (threadIdx.x < 16) {
    int n = threadIdx.x;
    const float* s = scores + b*256 + n*16;
    float mx = s[0];
    for (int m = 1; m < 16; m++) mx = fmaxf(mx, s[m]);
    float e[16]; float sum = 0.f;
    for (int m = 0; m < 16; m++) { e[m] = expf(s[m] - mx); sum += e[m]; }
    for (int m = 0; m < 16; m++) p[n][m] = e[m] / sum;
  }
  __syncthreads();
  const float* rb = rel + (size_t)b * (NN*DMM);
  for (int idx = threadIdx.x; idx < NN*DMM; idx += 256) {
    int n = idx >> 9, d = idx & 511;
    float s = 0.f;
    for (int m = 0; m < 16; m++) s += p[n][m] * rb[m*DMM + d];
    outcat[(size_t)b*16384 + n*1024 + d]       = s;
    outcat[(size_t)b*16384 + n*1024 + 512 + d] = rb[n*DMM + d];
  }
}

// ---------------------------------------------------------------------------
// Finalize: pad-mask scores, write attn_scores output, gumbel softmax,
// argmax -> straight-through w = (hard - y) + y   (computed literally).
// ---------------------------------------------------------------------------
__global__ void k_finalize(const float* __restrict__ sc16, const float* __restrict__ mask_sum,
                           const float* __restrict__ gum, float* __restrict__ out_scores,
                           float* __restrict__ w) {
  int b = blockIdx.x * blockDim.x + threadIdx.x;
  if (b >= BB) return;
  float s[16], e[16];
  float mx = -3.4e38f;
  for (int n = 0; n < 16; n++) {
    float v = sc16[b*16 + n];
    if (mask_sum[b*16 + n] == 0.f) v = -1e-06f;
    s[n] = v;
    float z = v + gum[b*16 + n];
    e[n] = z;
    mx = fmaxf(mx, z);
  }
  float sum = 0.f;
  for (int n = 0; n < 16; n++) { e[n] = expf(e[n] - mx); sum += e[n]; }
  int amax = 0; float best = -1.f;
  for (int n = 0; n < 16; n++) {
    float y = e[n] / sum; e[n] = y;
    if (y > best) { best = y; amax = n; }
  }
  for (int n = 0; n < 16; n++) {
    out_scores[b*16 + n] = s[n];
    float h = (n == amax) ? 1.f : 0.f;
    w[b*16 + n] = (h - e[n]) + e[n];
  }
}

// ---------------------------------------------------------------------------
// context[b,h,w] = sum_n w[b,n] * om[b,n,h,w].  Vectorized b128, and the om
// stream is its last use -> non-temporal to avoid evicting live L2 data.
// ---------------------------------------------------------------------------
__global__ void k_context(const float* __restrict__ om, const float* __restrict__ w,
                          float* __restrict__ ctx) {
  int idx = (int)(blockIdx.x * 256 + threadIdx.x);
  if (idx >= BB * HW4) return;
  int b = idx / HW4;
  int p = idx - b * HW4;
  const v4f* ob = (const v4f*)(om + (size_t)b * NN * HW) + p;
  const float* wb = w + b * 16;
  v4f acc = {};
#pragma unroll
  for (int n = 0; n < 16; n++) {
    v4f v = __builtin_nontemporal_load(ob + (size_t)n * HW4);
    float wv = wb[n];
    acc.x += wv * v.x; acc.y += wv * v.y; acc.z += wv * v.z; acc.w += wv * v.w;
  }
  __builtin_nontemporal_store(acc, (v4f*)ctx + idx);
}

// ---------------------------------------------------------------------------
// Host-side orchestration
// ---------------------------------------------------------------------------
static inline void gemm(const float* X, const float* W, const float* b, float* Y,
                        float* Yp, int K, int Nout, hipStream_t s) {
  int strips = Nout / 16;
  if (K > 4096) {
    // split-K: chunks of 2048, deterministic fixed-order reduction
    int nchunks = K / 2048;
    int waves = strips * nchunks;
    k_gemm_partial<<<(waves + 7) / 8, 256, 0, s>>>(X, W, Yp, K, Nout, 2048, nchunks);
    int total = 64 * Nout;
    k_reduce_bias<<<(total + 255) / 256, 256, 0, s>>>(Yp, b, Y, Nout, nchunks);
  } else {
    k_gemm_bias<<<(strips + 7) / 8, 256, 0, s>>>(X, W, b, Y, K, Nout);
  }
}
static inline void lnr(float* Y, const float* g, const float* be, int Ncols, hipStream_t s) {
  k_ln_relu<<<BB, 256, 0, s>>>(Y, g, be, Ncols);
}

extern "C" void kernel_launch(void* const* d_in, const int* in_sizes, int n_in,
                              void* d_out, int out_size, void* d_ws, size_t ws_size,
                              hipStream_t stream) {
  (void)in_sizes; (void)n_in; (void)out_size; (void)ws_size;
  // --- inputs (pytree leaves in setup_inputs dict order) ---
  const float* target_mask  = (const float*)d_in[0];   // (B,1,H,W)
  const float* object_masks = (const float*)d_in[1];   // (B,N,1,H,W)
  // d_in[2] raw_object_masks: unused (raw_context not returned)
  const float* bboxes       = (const float*)d_in[3];   // (B,N,4)
  const float* target_feats = (const float*)d_in[4];   // (B,1,D) -> 64x1024
  const float* object_feats = (const float*)d_in[5];   // (B,N,D) -> 64x16384
  const float* gumbel       = (const float*)d_in[6];   // (B,N)
  // rel_params leaves: 7..20 ; wt: 21..30 ; wo: 31..40 ; attn: 41..58
  const float** P = (const float**)d_in;

  float* ws = (float*)d_ws;
  float* Tsum   = ws + 0;            // 64
  float* msum   = ws + 64;           // 1024
  float* edge   = ws + 1088;         // 64*32
  float* wsel   = ws + 3136;         // 1024
  float* sc16   = ws + 4160;         // 64*16
  float* scores = ws + 5184;         // 64*256
  float* p0     = ws + 21568;        // 64*2048
  float* p1     = ws + 152640;       // 64*2048
  float* rel    = ws + 283712;       // 64*8192
  float* qb     = ws + 808000;       // 64*8192
  float* kb     = ws + 1332288;      // 64*8192
  float* outcat = ws + 1856576;      // 64*16384
  float* yp     = ws + 2905152;      // 8*64*2048 split-K partials
  // total: 3,953,728 floats = 15.8 MB

  float* ctx_out    = (float*)d_out;              // 3,211,264 floats
  float* scores_out = (float*)d_out + BB * HW;    // 1024 floats

  // 1) mask reductions + edge features
  k_tsum   <<<BB,      256, 0, stream>>>(target_mask, Tsum);
  k_overlap<<<BB*NN,   256, 0, stream>>>(object_masks, target_mask, edge, msum);
  k_iou    <<<BB*NN,   256, 0, stream>>>(target_mask, bboxes, Tsum, edge);

  // 2) rel MLP: 32 -> 512 -> 1024 -> 512 -> 8192
  gemm(edge, P[7],  P[8],  p0,  yp, 32,   512,  stream); lnr(p0, P[9],  P[10], 512,  stream);
  gemm(p0,   P[11], P[12], p1,  yp, 512,  1024, stream); lnr(p1, P[13], P[14], 1024, stream);
  gemm(p1,   P[15], P[16], p0,  yp, 1024, 512,  stream); lnr(p0, P[17], P[18], 512,  stream);
  gemm(p0,   P[19], P[20], rel, yp, 512,  8192, stream);

  // 3) wt MLP on target_feats: 1024 -> 2048 -> 1024 -> 8192
  gemm(target_feats, P[21], P[22], p0, yp, 1024, 2048, stream); lnr(p0, P[23], P[24], 2048, stream);
  gemm(p0,           P[25], P[26], p1, yp, 2048, 1024, stream); lnr(p1, P[27], P[28], 1024, stream);
  gemm(p1,           P[29], P[30], qb, yp, 1024, 8192, stream);

  // 4) wo MLP on object_feats: 16384 -> 2048 -> 1024 -> 8192 (split-K on L0)
  gemm(object_feats, P[31], P[32], p0, yp, 16384, 2048, stream); lnr(p0, P[33], P[34], 2048, stream);
  gemm(p0,           P[35], P[36], p1, yp, 2048,  1024, stream); lnr(p1, P[37], P[38], 1024, stream);
  gemm(p1,           P[39], P[40], kb, yp, 1024,  8192, stream);

  // 5) attention over N=16: WMMA scores, softmax + P@rel, concat
  k_scores_wmma<<<BB/8, 256, 0, stream>>>(qb, kb, scores);
  k_attn       <<<BB,   256, 0, stream>>>(scores, rel, outcat);

  // 6) attn MLP: 16384 -> 1024 -> 2048 -> 1024 -> 512 -> 16 (split-K on L0)
  gemm(outcat, P[41], P[42], p0,   yp, 16384, 1024, stream); lnr(p0, P[43], P[44], 1024, stream);
  gemm(p0,     P[45], P[46], p1,   yp, 1024,  2048, stream); lnr(p1, P[47], P[48], 2048, stream);
  gemm(p1,     P[49], P[50], p0,   yp, 2048,  1024, stream); lnr(p0, P[51], P[52], 1024, stream);
  gemm(p0,     P[53], P[54], p1,   yp, 1024,  512,  stream); lnr(p1, P[55], P[56], 512,  stream);
  gemm(p1,     P[57], P[58], sc16, yp, 512,   16,   stream);

  // 7) pad-mask + gumbel softmax + straight-through weights; write scores out
  k_finalize<<<(BB + 63) / 64, 64, 0, stream>>>(sc16, msum, gumbel, scores_out, wsel);

  // 8) context = sum_n w[b,n] * om[b,n,:,:]
  k_context<<<(BB * HW4 + 255) / 256, 256, 0, stream>>>(object_masks, wsel, ctx_out);
}